// RNNmodel_26525718020776
// MI455X (gfx1250) — compile-verified
//
#include <hip/hip_runtime.h>

typedef __attribute__((ext_vector_type(2))) float v2f;
typedef __attribute__((ext_vector_type(4))) float v4f;
typedef __attribute__((ext_vector_type(8))) float v8f;

namespace {
constexpr int kB = 2048;
constexpr int kT = 1024;
constexpr int kH = 20;           // 3H = 60, padded to 64 rows for 4 WMMA M-tiles
constexpr int kLdsStride = 68;   // padded row stride (floats): conflict-free b64 loads, 16B-aligned b128 stores

__device__ __forceinline__ float fast_sigmoid(float x) {
  // sigmoid(x) = 1 / (1 + 2^(-x*log2(e)))
  float e = __builtin_amdgcn_exp2f(x * -1.44269504088896340736f);
  return __builtin_amdgcn_rcpf(1.0f + e);
}

__device__ __forceinline__ float fast_tanh(float x) {
#if __has_builtin(__builtin_amdgcn_tanhf)
  return __builtin_amdgcn_tanhf(x);
#else
  // tanh(x) = 1 - 2/(2^(2x*log2(e)) + 1)
  float e = __builtin_amdgcn_exp2f(x * 2.88539008177792681472f);
  return 1.0f - 2.0f * __builtin_amdgcn_rcpf(e + 1.0f);
#endif
}

#define LDS_FENCE() asm volatile("s_wait_dscnt 0" ::: "memory")
} // namespace

// One wave32 per 16-row batch tile. Per step:
//   gh^T[64x16] = W_hh[64x20] @ h^T[20x16] + b_hh   via 20x V_WMMA_F32_16X16X4_F32
// A (W_hh) operands are loop-invariant registers; B (h^T) operands are
// regenerated in-register by the gate math each step; D is transposed
// through a tiny padded LDS slice.
__global__ __launch_bounds__(32) void gru_wmma_f32(
    const float* __restrict__ X,      // [B, T, 1]
    const float* __restrict__ h0,     // [1, B, H]
    const float* __restrict__ W_ih,   // [3H, 1]
    const float* __restrict__ W_hh,   // [3H, H]
    const float* __restrict__ b_ih,   // [3H]
    const float* __restrict__ b_hh,   // [3H]
    const float* __restrict__ W_out,  // [1, H]
    const float* __restrict__ b_out,  // [1]
    float* __restrict__ out)          // [B*T] y  ++  [B*H] h_last
{
  __shared__ float lds[16 * kLdsStride];

  const int lane = threadIdx.x & 31;
  const int bl   = lane & 15;        // batch column within tile (WMMA N / B-matrix lane)
  const int hi   = lane >> 4;        // half-wave selector (K pairs 0,1 vs 2,3)
  const int bg   = blockIdx.x * 16 + bl;  // global batch row
  float* wrow = &lds[bl * kLdsStride];

  // ---- Loop-invariant A operands: W_hh as 4 (Mtile) x 5 (Kchunk) 16x4 tiles.
  // A layout: lanes 0-15 hold M=lane, K=4c+{0,1}; lanes 16-31 hold K=4c+{2,3}.
  v2f a[4][5];
#pragma unroll
  for (int m = 0; m < 4; ++m) {
    const int g = 16 * m + bl;       // M row = gate row (padded 60->64 with zeros)
#pragma unroll
    for (int c = 0; c < 5; ++c) {
      const int k = 4 * c + 2 * hi;
      v2f t;
      t.x = (g < 60) ? W_hh[g * kH + k]     : 0.0f;
      t.y = (g < 60) ? W_hh[g * kH + k + 1] : 0.0f;
      a[m][c] = t;
    }
  }

  // ---- Accumulator init = b_hh broadcast (D layout: row g = 16m + v + 8*hi, col = lane%16)
  v8f cinit[4];
#pragma unroll
  for (int m = 0; m < 4; ++m) {
#pragma unroll
    for (int v = 0; v < 8; ++v) {
      const int g = 16 * m + 8 * hi + v;
      cinit[m][v] = (g < 60) ? b_hh[g] : 0.0f;
    }
  }

  // ---- Per-slot input-projection coefficients. Slot s handles hidden index
  // j = 4*(s/2) + (s&1) + 2*hi, matching the B-operand layout exactly.
  float wr[10], wz[10], wn[10], br[10], bz[10], bn[10], wo[10];
#pragma unroll
  for (int s = 0; s < 10; ++s) {
    const int j = 4 * (s >> 1) + (s & 1) + 2 * hi;
    wr[s] = W_ih[j];       wz[s] = W_ih[j + 20];  wn[s] = W_ih[j + 40];
    br[s] = b_ih[j];       bz[s] = b_ih[j + 20];  bn[s] = b_ih[j + 40];
    wo[s] = W_out[j];
  }
  const float bo = b_out[0];

  // ---- Initial hidden state directly in B-operand layout.
  v2f hB[5];
#pragma unroll
  for (int c = 0; c < 5; ++c) {
    const int k = 4 * c + 2 * hi;
    v2f t;
    t.x = h0[bg * kH + k];
    t.y = h0[bg * kH + k + 1];
    hB[c] = t;
  }

  const float* Xp = X + (size_t)bg * kT;
  float* Y  = out;
  float* HL = out + (size_t)kB * kT;

  v4f xa = *(const v4f*)(Xp);   // double-buffered X (4 steps per lane)

  for (int t0 = 0; t0 < kT; t0 += 4) {
    v4f xn;
    xn.x = xn.y = xn.z = xn.w = 0.0f;
    if (t0 + 4 < kT) xn = *(const v4f*)(Xp + t0 + 4);

    v4f yb;
#pragma unroll
    for (int u = 0; u < 4; ++u) {
      const float x = xa[u];

      // ---- gh^T = W_hh @ h^T + b_hh : 4 independent chains of 5 chained WMMAs
      v8f acc[4];
#pragma unroll
      for (int m = 0; m < 4; ++m)
        acc[m] = __builtin_amdgcn_wmma_f32_16x16x4_f32(
            false, a[m][0], false, hB[0], (short)0, cinit[m], false, false);
#pragma unroll
      for (int c = 1; c < 5; ++c) {
#pragma unroll
        for (int m = 0; m < 4; ++m)
          acc[m] = __builtin_amdgcn_wmma_f32_16x16x4_f32(
              false, a[m][c], false, hB[c], (short)0, acc[m], false, false);
      }

      // ---- Transpose D through LDS: lane owns column b, writes rows 16m+8hi+0..7
#pragma unroll
      for (int m = 0; m < 4; ++m) {
        v4f lo, hi4;
        lo.x  = acc[m][0]; lo.y  = acc[m][1]; lo.z  = acc[m][2]; lo.w  = acc[m][3];
        hi4.x = acc[m][4]; hi4.y = acc[m][5]; hi4.z = acc[m][6]; hi4.w = acc[m][7];
        *(v4f*)(wrow + 16 * m + 8 * hi)     = lo;
        *(v4f*)(wrow + 16 * m + 8 * hi + 4) = hi4;
      }
      LDS_FENCE();  // same-wave in-order DS + data-count wait before readback

      // ---- Gate math; h_new written straight back into B-operand registers
      float yp = 0.0f;
#pragma unroll
      for (int c = 0; c < 5; ++c) {
        const int j0 = 4 * c + 2 * hi;
        const v2f gr = *(const v2f*)(wrow + j0);        // gh_r rows j0, j0+1
        const v2f gz = *(const v2f*)(wrow + j0 + 20);   // gh_z
        const v2f gn = *(const v2f*)(wrow + j0 + 40);   // gh_n (includes b_hh via cinit)
        const int s0 = 2 * c, s1 = 2 * c + 1;
        const float hx = hB[c].x, hy = hB[c].y;

        float r0 = fast_sigmoid(fmaf(x, wr[s0], br[s0]) + gr.x);
        float z0 = fast_sigmoid(fmaf(x, wz[s0], bz[s0]) + gz.x);
        float n0 = fast_tanh(fmaf(x, wn[s0], bn[s0]) + r0 * gn.x);
        float hn0 = n0 + z0 * (hx - n0);                // (1-z)n + z*h

        float r1 = fast_sigmoid(fmaf(x, wr[s1], br[s1]) + gr.y);
        float z1 = fast_sigmoid(fmaf(x, wz[s1], bz[s1]) + gz.y);
        float n1 = fast_tanh(fmaf(x, wn[s1], bn[s1]) + r1 * gn.y);
        float hn1 = n1 + z1 * (hy - n1);

        v2f t; t.x = hn0; t.y = hn1;
        hB[c] = t;
        yp = fmaf(hn0, wo[s0], yp);
        yp = fmaf(hn1, wo[s1], yp);
      }

      // ---- y[b,t] = h_new . W_out + b_out (reduce across the lane pair)
      float ys = yp + __shfl_xor(yp, 16, 32);
      yb[u] = ys + bo;
    }

    if (hi == 0) *(v4f*)(Y + (size_t)bg * kT + t0) = yb;
    xa = xn;
  }

  // ---- h_last [B, H]
#pragma unroll
  for (int c = 0; c < 5; ++c) {
    const int k = 4 * c + 2 * hi;
    *(v2f*)(HL + (size_t)bg * kH + k) = hB[c];
  }
}

extern "C" void kernel_launch(void* const* d_in, const int* in_sizes, int n_in,
                              void* d_out, int out_size, void* d_ws, size_t ws_size,
                              hipStream_t stream) {
  (void)in_sizes; (void)n_in; (void)out_size; (void)d_ws; (void)ws_size;
  const float* X     = (const float*)d_in[0];
  const float* h0    = (const float*)d_in[1];
  const float* W_ih  = (const float*)d_in[2];
  const float* W_hh  = (const float*)d_in[3];
  const float* b_ih  = (const float*)d_in[4];
  const float* b_hh  = (const float*)d_in[5];
  const float* W_out = (const float*)d_in[6];
  const float* b_out = (const float*)d_in[7];
  float* out = (float*)d_out;

  dim3 grid(kB / 16);   // 128 single-wave tiles -> spread across WGPs
  dim3 block(32);       // one wave32 per workgroup, no barriers needed
  gru_wmma_f32<<<grid, block, 0, stream>>>(X, h0, W_ih, W_hh, b_ih, b_hh,
                                           W_out, b_out, out);
}